// GcnNet_17695265259748
// MI455X (gfx1250) — compile-verified
//
#include <hip/hip_runtime.h>
#include <hip/hip_bf16.h>

// ---------------------------------------------------------------------------
// GCN layer for MI455X (gfx1250, wave32).
//   h   = (mean_L x) @ W + b            -- fp32 WMMA 16x16x4
//   out = D^-1/2 (A+I) D^-1/2 h         -- L2-resident gather + f32 atomics
// ---------------------------------------------------------------------------

typedef __attribute__((ext_vector_type(2))) float v2f;
typedef __attribute__((ext_vector_type(8))) float v8f;

#define L_TOK 20
#define C_IN  128
#define F_OUT 300

// ---- 1) xbar[n][c] = mean over L of x[n][l][c] -----------------------------
__global__ __launch_bounds__(128) void mean_kernel(const float* __restrict__ x,
                                                   float* __restrict__ xbar) {
  const int n = blockIdx.x;
  const int c = threadIdx.x;            // 0..127, coalesced over channel dim
  const float* p = x + (size_t)n * (L_TOK * C_IN) + c;
  float s = 0.0f;
#pragma unroll
  for (int l = 0; l < L_TOK; ++l) s += p[l * C_IN];
  xbar[(size_t)n * C_IN + c] = s * (1.0f / (float)L_TOK);
}

// ---- 2) h = xbar @ W + b  via V_WMMA_F32_16X16X4_F32 -----------------------
// One wave computes one 16x16 tile of h. K = 128 -> 32 WMMA steps.
// A layout (16x4 f32): lane l, vgpr v -> (m = l&15, k = 2*(l>>4) + v)
// B layout (4x16 f32): lane l, vgpr v -> (k = 2*(l>>4) + v, n = l&15)
// C/D layout: vgpr r, lane l -> (m = r + 8*(l>>4), n = l&15)
__global__ __launch_bounds__(256) void gemm_wmma_kernel(
    const float* __restrict__ xbar,   // [N,128]
    const float* __restrict__ W,      // [128,300]
    const float* __restrict__ bias,   // [300]
    float* __restrict__ h,            // [N,300]
    int N, int tilesN, int totalTiles) {
  const int waveId = (int)((blockIdx.x * blockDim.x + threadIdx.x) >> 5);
  const int lane   = (int)(threadIdx.x & 31);
  if (waveId >= totalTiles) return;   // wave-uniform: EXEC stays all-1s for WMMA

  const int tm   = waveId / tilesN;
  const int tn   = waveId % tilesN;
  const int row0 = tm << 4;
  const int col0 = tn << 4;
  const int m    = lane & 15;
  const int kh   = lane >> 4;         // 0 or 1

  int arow = row0 + m;  if (arow > N - 1) arow = N - 1;          // clamp (store is guarded)
  const int bcol  = col0 + m;
  int bcolc = bcol; if (bcolc > F_OUT - 1) bcolc = F_OUT - 1;    // clamp (store is guarded)

  const float* aptr = xbar + (size_t)arow * C_IN + 2 * kh;
  const float* bptr = W + (size_t)(2 * kh) * F_OUT + bcolc;

  v8f acc = {};
#pragma unroll
  for (int kb = 0; kb < C_IN; kb += 4) {
    v2f a = *(const v2f*)(aptr + kb);           // xbar[arow][kb+2kh .. +1]
    v2f bb;
    bb.x = bptr[(size_t)kb * F_OUT];            // W[kb+2kh  ][bcol]
    bb.y = bptr[(size_t)(kb + 1) * F_OUT];      // W[kb+2kh+1][bcol]
    acc = __builtin_amdgcn_wmma_f32_16x16x4_f32(
        /*neg_a=*/false, a, /*neg_b=*/false, bb,
        /*c_mod=*/(short)0, acc, /*reuse_a=*/false, /*reuse_b=*/false);
  }

  if (bcol < F_OUT) {
    const float bv   = bias[bcol];
    const int mbase  = row0 + 8 * kh;
#pragma unroll
    for (int r = 0; r < 8; ++r) {
      const int rr = mbase + r;
      if (rr < N) h[(size_t)rr * F_OUT + bcol] = acc[r] + bv;
    }
  }
}

// ---- 3) degree (self loop included) ----------------------------------------
__global__ void deg_init_kernel(unsigned* __restrict__ deg, int N) {
  const int i = blockIdx.x * blockDim.x + threadIdx.x;
  if (i < N) deg[i] = 1u;             // self loop
}

__global__ void deg_count_kernel(const long long* __restrict__ ei,
                                 unsigned* __restrict__ deg, int E) {
  const int e = blockIdx.x * blockDim.x + threadIdx.x;
  if (e >= E) return;
  const long long r = ei[e];
  const long long c = ei[(size_t)E + e];
  if (r != c) atomicAdd(&deg[(int)r], 1u);
}

__global__ void dis_kernel(const unsigned* __restrict__ deg,
                           float* __restrict__ dis, int N) {
  const int i = blockIdx.x * blockDim.x + threadIdx.x;
  if (i < N) dis[i] = rsqrtf((float)deg[i]);
}

// ---- 4) self-loop term initializes out: out[n][:] = dis[n]^2 * h[n][:] -----
__global__ void selfloop_kernel(const float* __restrict__ h,
                                const float* __restrict__ dis,
                                float* __restrict__ out, int total) {
  const int idx = blockIdx.x * blockDim.x + threadIdx.x;
  if (idx >= total) return;
  const int n = idx / F_OUT;
  const float d = dis[n];
  out[idx] = d * d * h[idx];
}

// ---- 5) edge messages: one wave32 per edge, lanes stride features ----------
__global__ __launch_bounds__(256) void edge_scatter_kernel(
    const long long* __restrict__ ei,
    const float* __restrict__ dis,
    const float* __restrict__ h,
    float* __restrict__ out, int E) {
  const int wave = (int)((blockIdx.x * blockDim.x + threadIdx.x) >> 5);
  const int lane = (int)(threadIdx.x & 31);
  if (wave >= E) return;
  const long long r = ei[wave];
  const long long c = ei[(size_t)E + wave];
  if (r == c) return;                 // masked, like the reference
  const float norm = dis[(int)r] * dis[(int)c];
  const float* hs = h + (size_t)r * F_OUT;
  float* od       = out + (size_t)c * F_OUT;
  for (int j = lane; j < F_OUT; j += 32)
    atomicAdd(&od[j], norm * hs[j]); // non-returning f32 atomic, L2-resident
}

// ---------------------------------------------------------------------------
extern "C" void kernel_launch(void* const* d_in, const int* in_sizes, int n_in,
                              void* d_out, int out_size, void* d_ws, size_t ws_size,
                              hipStream_t stream) {
  const float*     x  = (const float*)d_in[0];       // [N,20,128] f32
  const long long* ei = (const long long*)d_in[1];   // [2,E] int64 (row then col)
  const float*     W  = (const float*)d_in[2];       // [128,300] f32
  const float*     b  = (const float*)d_in[3];       // [300] f32
  float* out = (float*)d_out;                        // [N,300] f32

  const int N = in_sizes[0] / (L_TOK * C_IN);
  const int E = in_sizes[1] / 2;

  // workspace layout (256B-aligned slices)
  auto align256 = [](size_t v) { return (v + 255) & ~(size_t)255; };
  char* ws = (char*)d_ws;
  size_t off = 0;
  float* xbar = (float*)(ws + off); off += align256((size_t)N * C_IN * sizeof(float));
  float* h    = (float*)(ws + off); off += align256((size_t)N * F_OUT * sizeof(float));
  unsigned* deg = (unsigned*)(ws + off); off += align256((size_t)N * sizeof(unsigned));
  float* dis  = (float*)(ws + off);

  // 1) token-mean (512 MB HBM read -> memory bound)
  mean_kernel<<<N, 128, 0, stream>>>(x, xbar);

  // 2) WMMA GEMM: tiles = ceil(N/16) x ceil(300/16)
  const int tilesM = (N + 15) >> 4;
  const int tilesN = (F_OUT + 15) >> 4;            // 19
  const int totalTiles = tilesM * tilesN;
  const int gemmBlocks = (totalTiles + 7) / 8;     // 8 waves / 256-thread block
  gemm_wmma_kernel<<<gemmBlocks, 256, 0, stream>>>(xbar, W, b, h, N, tilesN, totalTiles);

  // 3) degrees + dis = deg^-1/2
  deg_init_kernel<<<(N + 255) / 256, 256, 0, stream>>>(deg, N);
  deg_count_kernel<<<(E + 255) / 256, 256, 0, stream>>>(ei, deg, E);
  dis_kernel<<<(N + 255) / 256, 256, 0, stream>>>(deg, dis, N);

  // 4) self-loop contribution (also initializes every element of d_out)
  const int total = N * F_OUT;
  selfloop_kernel<<<(total + 255) / 256, 256, 0, stream>>>(h, dis, out, total);

  // 5) edge scatter: one wave per edge, f32 atomics into L2-resident out
  const int edgeBlocks = (E + 7) / 8;              // 8 waves per block
  edge_scatter_kernel<<<edgeBlocks, 256, 0, stream>>>(ei, dis, h, out, E);
}